// dQPEq_35759897707193
// MI455X (gfx1250) — compile-verified
//
#include <hip/hip_runtime.h>
#include <hip/hip_bf16.h>
#include <math.h>

// ---------------------------------------------------------------------------
// dQPEq: 40-iter primal-dual IPM.  nx=4096, neq=768, Q = 0.1*I.
// Heavy op: ADA = (A*diag(Dinv)) @ A^T  (768x768x4096)  ->  V_WMMA_F32_16X16X4_F32
// with K-blocked (256) promotion into f64 accumulators (mixed-precision
// inexact Newton; residuals stay f64).  Everything is L2-resident (~55MB
// working set vs 192MB L2), so HBM BW is irrelevant after first touch.
// ---------------------------------------------------------------------------

#define NXV  4096
#define NEQV 768
#define MUQ  0.1
#define SIGC 0.1
#define ITERS 40

typedef __attribute__((ext_vector_type(2))) float v2f;
typedef __attribute__((ext_vector_type(8))) float v8f;

// -------------------------------- setup ------------------------------------
__global__ void k_init(const float* __restrict__ puzzles,
                       const double* __restrict__ log_z0,
                       double* q, double* z, double* lam, double* nu, double* ez) {
    int i = blockIdx.x * blockDim.x + threadIdx.x;
    if (i < NXV) {
        q[i]   = -(double)puzzles[i];
        z[i]   = 1.0;
        lam[i] = 1.0;
        ez[i]  = exp(log_z0[i]);
    }
    if (i < NEQV) nu[i] = 0.0;
}

__global__ void k_convertA(const double* __restrict__ A, float* __restrict__ Af32) {
    size_t idx = (size_t)blockIdx.x * blockDim.x + threadIdx.x;
    Af32[idx] = (float)A[idx];
}

// out[row] = A@x + beta*axpy[row]    (wave per row, wave32 shuffle reduce)
__global__ void k_gemv_A(const double* __restrict__ A, const double* __restrict__ x,
                         const double* __restrict__ axpy, double beta,
                         double* __restrict__ out) {
    int wave = threadIdx.x >> 5, lane = threadIdx.x & 31;
    int row  = blockIdx.x * 8 + wave;                 // 96 blocks * 8 waves = 768
    const double* ar = A + (size_t)row * NXV;
    double s = 0.0;
    for (int k = lane; k < NXV; k += 32) s += ar[k] * x[k];
    for (int off = 16; off; off >>= 1) s += __shfl_down(s, off, 32);
    if (lane == 0) out[row] = s + beta * axpy[row];
}

// out[col] = A^T @ y   (coalesced row-major sweep; y is SGPR-uniform)
__global__ void k_gemv_AT(const double* __restrict__ A, const double* __restrict__ y,
                          double* __restrict__ out) {
    int col = blockIdx.x * blockDim.x + threadIdx.x;  // 4096
    double s = 0.0;
    for (int i = 0; i < NEQV; ++i) s += A[(size_t)i * NXV + col] * y[i];
    out[col] = s;
}

// gap, rd, rc, Dinv, rtil, w = rtil*Dinv    (one workgroup, LDS reduction)
__global__ void __launch_bounds__(1024)
k_center(const double* __restrict__ q, const double* __restrict__ z,
         const double* __restrict__ lam, const double* __restrict__ t1,
         double* Dinv, double* rc, double* rtil, double* w) {
    __shared__ double red[1024];
    __shared__ double gapS;
    int tid = threadIdx.x;
    double s = 0.0;
    for (int i = tid; i < NXV; i += 1024) s += z[i] * lam[i];
    red[tid] = s; __syncthreads();
    for (int h = 512; h; h >>= 1) { if (tid < h) red[tid] += red[tid + h]; __syncthreads(); }
    if (tid == 0) gapS = red[0] / (double)NXV;
    __syncthreads();
    double gap = gapS;
    for (int i = tid; i < NXV; i += 1024) {
        double zi = z[i], li = lam[i];
        double rd  = MUQ * zi + q[i] - li + t1[i];
        double rci = zi * li - SIGC * gap;
        double di  = 1.0 / (MUQ + li / zi);
        double rt  = -rd - rci / zi;
        rc[i] = rci; Dinv[i] = di; rtil[i] = rt; w[i] = rt * di;
    }
}

// Wf32[i,k] = f32( A[i,k] * Dinv[k] )
__global__ void k_scaleW(const double* __restrict__ A, const double* __restrict__ Dinv,
                         float* __restrict__ Wf32) {
    size_t idx = (size_t)blockIdx.x * blockDim.x + threadIdx.x;
    Wf32[idx] = (float)(A[idx] * Dinv[idx & (NXV - 1)]);
}

// ------------------- WMMA Schur GEMM:  M = Wf32 @ Af32^T -------------------
// 64x64 block tile, BK=32, 8 waves: wave (wr,wc) computes 16 rows x 32 cols
// (two 16x16 accumulators).  LDS stride 34 floats -> conflict-free b64 reads.
#define LDST 34
__global__ void __launch_bounds__(256)
k_schur_wmma(const float* __restrict__ Wf32, const float* __restrict__ Af32,
             double* __restrict__ Mout) {
    __shared__ float sW[64 * LDST];
    __shared__ float sA[64 * LDST];
    const int tid  = threadIdx.x;
    const int wave = tid >> 5, lane = tid & 31;
    const int bx = blockIdx.x % 12, by = blockIdx.x / 12;   // 12x12 = 144 blocks
    const int rowBase = by * 64, colBase = bx * 64;
    const int wr = (wave & 3) * 16;     // row sub-tile
    const int wc = (wave >> 2) * 32;    // col sub-tile (0 or 32)
    const int r  = lane & 15, h = lane >> 4;

    v8f c0 = {0.f,0.f,0.f,0.f,0.f,0.f,0.f,0.f};
    v8f c1 = c0;
    double d0[8] = {0}, d1[8] = {0};

    const int lrow = tid >> 3;          // 0..31
    const int lk   = (tid & 7) * 4;     // 0..28

    for (int k0 = 0; k0 < NXV; k0 += 32) {
        __syncthreads();
        for (int rr = 0; rr < 64; rr += 32) {
            const float4 wv = *(const float4*)(Wf32 + (size_t)(rowBase + lrow + rr) * NXV + k0 + lk);
            const float4 av = *(const float4*)(Af32 + (size_t)(colBase + lrow + rr) * NXV + k0 + lk);
            int sb = (lrow + rr) * LDST + lk;
            sW[sb+0] = wv.x; sW[sb+1] = wv.y; sW[sb+2] = wv.z; sW[sb+3] = wv.w;
            sA[sb+0] = av.x; sA[sb+1] = av.y; sA[sb+2] = av.z; sA[sb+3] = av.w;
        }
        __syncthreads();
#pragma unroll
        for (int kk = 0; kk < 32; kk += 4) {
            v2f a  = *(const v2f*)(sW + (wr + r)      * LDST + kk + 2 * h);
            v2f b0 = *(const v2f*)(sA + (wc + r)      * LDST + kk + 2 * h);
            v2f b1 = *(const v2f*)(sA + (wc + 16 + r) * LDST + kk + 2 * h);
            c0 = __builtin_amdgcn_wmma_f32_16x16x4_f32(false, a, false, b0, (short)0, c0, false, false);
            c1 = __builtin_amdgcn_wmma_f32_16x16x4_f32(false, a, false, b1, (short)0, c1, false, false);
        }
        if (((k0 >> 5) & 7) == 7) {     // promote every 256 K into f64
#pragma unroll
            for (int j = 0; j < 8; ++j) {
                d0[j] += (double)c0[j]; c0[j] = 0.f;
                d1[j] += (double)c1[j]; c1[j] = 0.f;
            }
        }
    }
    const int gr = rowBase + wr + 8 * h;
    const int gc = colBase + wc + r;
#pragma unroll
    for (int j = 0; j < 8; ++j) {
        Mout[(size_t)(gr + j) * NEQV + gc]      = d0[j];
        Mout[(size_t)(gr + j) * NEQV + gc + 16] = d1[j];
    }
}

// ------------- Cholesky (in place, lower) + two triangular solves ----------
__global__ void __launch_bounds__(1024)
k_cholsolve(double* __restrict__ M, const double* __restrict__ rhs,
            double* __restrict__ dnu) {
    const int n = NEQV, NT = 1024;
    int tid = threadIdx.x;
    __shared__ double r[NEQV], yv[NEQV];
    for (int i = tid; i < n; i += NT) r[i] = rhs[i];
    __syncthreads();
    // right-looking Cholesky (lower triangle only)
    for (int j = 0; j < n; ++j) {
        if (tid == 0) M[(size_t)j * n + j] = sqrt(M[(size_t)j * n + j]);
        __syncthreads();
        double inv = 1.0 / M[(size_t)j * n + j];
        for (int i = j + 1 + tid; i < n; i += NT) M[(size_t)i * n + j] *= inv;
        __syncthreads();
        for (int i = j + 1 + tid; i < n; i += NT) {
            double lij = M[(size_t)i * n + j];
            double* ri = M + (size_t)i * n;
            for (int k = j + 1; k <= i; ++k) ri[k] -= lij * M[(size_t)k * n + j];
        }
        __syncthreads();
    }
    // forward: L y = rhs (column-oriented)
    for (int j = 0; j < n; ++j) {
        double yj = r[j] / M[(size_t)j * n + j];
        if (tid == 0) yv[j] = yj;
        for (int i = j + 1 + tid; i < n; i += NT) r[i] -= M[(size_t)i * n + j] * yj;
        __syncthreads();
    }
    // backward: L^T x = y  (reads row j of L -> coalesced)
    for (int j = n - 1; j >= 0; --j) {
        double xj = yv[j] / M[(size_t)j * n + j];
        if (tid == 0) dnu[j] = xj;
        for (int i = tid; i < j; i += NT) yv[i] -= M[(size_t)j * n + i] * xj;
        __syncthreads();
    }
}

// ------------- dz, dlam, fraction-to-boundary, state update ----------------
__global__ void __launch_bounds__(1024)
k_step(const double* __restrict__ rtil, const double* __restrict__ t2,
       const double* __restrict__ Dinv, const double* __restrict__ rc,
       double* z, double* lam, double* nu, const double* __restrict__ dnu) {
    __shared__ double red[1024];
    __shared__ double alphaS;
    int tid = threadIdx.x;
    double dz[4], dl[4];
    double am = 1e300;
#pragma unroll
    for (int kk = 0; kk < 4; ++kk) {
        int i = tid + kk * 1024;
        double zi = z[i], li = lam[i];
        double d = (rtil[i] - t2[i]) * Dinv[i];
        double l = (-rc[i] - li * d) / zi;
        dz[kk] = d; dl[kk] = l;
        if (d < 0.0) am = fmin(am, -zi / d);
        if (l < 0.0) am = fmin(am, -li / l);
    }
    red[tid] = am; __syncthreads();
    for (int h = 512; h; h >>= 1) { if (tid < h) red[tid] = fmin(red[tid], red[tid + h]); __syncthreads(); }
    if (tid == 0) alphaS = fmin(1.0, 0.99 * red[0]);
    __syncthreads();
    double a = alphaS;
#pragma unroll
    for (int kk = 0; kk < 4; ++kk) {
        int i = tid + kk * 1024;
        z[i]   += a * dz[kk];
        lam[i] += a * dl[kk];
    }
    if (tid < NEQV) nu[tid] += a * dnu[tid];
}

__global__ void k_out(const double* __restrict__ z, float* __restrict__ out) {
    int i = blockIdx.x * blockDim.x + threadIdx.x;
    if (i < NXV) out[i] = (float)z[i];
}

// ---------------------------------------------------------------------------
extern "C" void kernel_launch(void* const* d_in, const int* in_sizes, int n_in,
                              void* d_out, int out_size, void* d_ws, size_t ws_size,
                              hipStream_t stream) {
    const float*  puzzles = (const float*)d_in[0];
    const double* A       = (const double*)d_in[1];
    const double* log_z0  = (const double*)d_in[2];
    float* out = (float*)d_out;

    // workspace carve-up (~31 MB total; all rewritten every call)
    size_t o = 0;
    auto alloc_d = [&](size_t n) { double* p = (double*)((char*)d_ws + o); o += n * 8; return p; };
    double* q    = alloc_d(NXV);
    double* z    = alloc_d(NXV);
    double* lam  = alloc_d(NXV);
    double* Dinv = alloc_d(NXV);
    double* rc   = alloc_d(NXV);
    double* rtil = alloc_d(NXV);
    double* w    = alloc_d(NXV);
    double* tvec = alloc_d(NXV);      // A^T nu, later A^T dnu
    double* ez   = alloc_d(NXV);
    double* nu   = alloc_d(NEQV);
    double* b    = alloc_d(NEQV);
    double* rp   = alloc_d(NEQV);
    double* rhs  = alloc_d(NEQV);
    double* dnu  = alloc_d(NEQV);
    double* Mmat = alloc_d((size_t)NEQV * NEQV);
    o = (o + 15) & ~(size_t)15;
    float* Af32 = (float*)((char*)d_ws + o); o += (size_t)NEQV * NXV * 4;
    float* Wf32 = (float*)((char*)d_ws + o); o += (size_t)NEQV * NXV * 4;

    const int elemBlocks = (NEQV * NXV) / 256;   // 12288

    k_init<<<NXV / 256, 256, 0, stream>>>(puzzles, log_z0, q, z, lam, nu, ez);
    k_convertA<<<elemBlocks, 256, 0, stream>>>(A, Af32);
    k_gemv_A<<<96, 256, 0, stream>>>(A, ez, ez, 0.0, b);          // b = A exp(log_z0)

    for (int it = 0; it < ITERS; ++it) {
        k_gemv_AT<<<16, 256, 0, stream>>>(A, nu, tvec);           // t1 = A^T nu
        k_gemv_A<<<96, 256, 0, stream>>>(A, z, b, -1.0, rp);      // rp = A z - b
        k_center<<<1, 1024, 0, stream>>>(q, z, lam, tvec, Dinv, rc, rtil, w);
        k_scaleW<<<elemBlocks, 256, 0, stream>>>(A, Dinv, Wf32);  // W = A*diag(Dinv) (f32)
        k_gemv_A<<<96, 256, 0, stream>>>(A, w, rp, 1.0, rhs);     // rhs = A(rtil*Dinv)+rp
        k_schur_wmma<<<144, 256, 0, stream>>>(Wf32, Af32, Mmat);  // ADA via WMMA
        k_cholsolve<<<1, 1024, 0, stream>>>(Mmat, rhs, dnu);
        k_gemv_AT<<<16, 256, 0, stream>>>(A, dnu, tvec);          // t2 = A^T dnu
        k_step<<<1, 1024, 0, stream>>>(rtil, tvec, Dinv, rc, z, lam, nu, dnu);
    }
    k_out<<<NXV / 256, 256, 0, stream>>>(z, out);
}